// InputProcessor_16475494548017
// MI455X (gfx1250) — compile-verified
//
#include <hip/hip_runtime.h>
#include <stdint.h>

// Problem shape (from the reference): x:(1024,2048) int32 -> out:(1024,32128) f32
#define BATCH     1024
#define SEQ       2048
#define VOCAB     32128
#define VOCAB_PAD 32768   // pad to 128*256 so the zero loop is uniform (32 float4/thread)
#define TPB       256

typedef __attribute__((ext_vector_type(4))) uint32_t v4u;   // D# group 0 (4 SGPRs)
typedef __attribute__((ext_vector_type(8))) uint32_t v8u;   // D# group 1 (8 SGPRs)

// One workgroup per batch row.
//  1) zero a padded f32 histogram in LDS (131 KB -- fits CDNA5's 320KB WGP LDS, 2 WGs/WGP)
//  2) ds_add_f32 scatter of the row's 2048 tokens (accumulate directly in f32)
//  3) single TDM tensor_store_from_lds DMAs the finished 125.5 KB row LDS -> HBM
__global__ __launch_bounds__(TPB) void row_hist_tdm_kernel(const int* __restrict__ x,
                                                           float* __restrict__ out) {
    __shared__ __attribute__((aligned(16))) float hist[VOCAB_PAD];

    const int tid = threadIdx.x;
    const int b   = blockIdx.x;

    // ---- 1) zero LDS histogram: uniform trip count, straight-line ds_store_b128 ----
    float4* h4 = reinterpret_cast<float4*>(hist);
    const float4 z = make_float4(0.f, 0.f, 0.f, 0.f);
    #pragma unroll
    for (int i = 0; i < VOCAB_PAD / (4 * TPB); ++i)   // exactly 32 stores per thread
        h4[i * TPB + tid] = z;
    __syncthreads();

    // ---- 2) scatter-add this row's tokens into LDS (ds_add_f32, no-return) ----
    const int4* xr = reinterpret_cast<const int4*>(x + (size_t)b * SEQ);
    #pragma unroll
    for (int it = 0; it < SEQ / (TPB * 4); ++it) {    // 2 coalesced int4 loads/thread
        int4 t = xr[it * TPB + tid];
        atomicAdd(&hist[t.x], 1.0f);
        atomicAdd(&hist[t.y], 1.0f);
        atomicAdd(&hist[t.z], 1.0f);
        atomicAdd(&hist[t.w], 1.0f);
    }
    __syncthreads();

    // ---- 3) TDM: DMA the 32128 real elements from LDS to global in one op ----
    if (tid == 0) {
        const uint64_t ga  = (uint64_t)(uintptr_t)(out + (size_t)b * VOCAB);
        const uint32_t lds = (uint32_t)(uintptr_t)(&hist[0]);   // LDS byte offset

        v4u g0;  // D# group 0: control / lds_addr / global_addr / type
        g0[0] = 1u;                                  // count=1, is_restore=0, no gather
        g0[1] = lds;                                 // lds_addr (bytes)
        g0[2] = (uint32_t)ga;                        // global_addr[31:0]
        g0[3] = (uint32_t)(ga >> 32) | (2u << 30);   // global_addr[56:32] | type=2

        v8u g1;  // D# group 1: data_size / dims / tile / stride
        g1[0] = 2u << 16;                            // workgroup_mask=0, data_size=2 (4B)
        g1[1] = (uint32_t)(VOCAB & 0xFFFF) << 16;    // tensor_dim0[15:0]  @ bits 63:48
        g1[2] = (uint32_t)(VOCAB >> 16) | (1u << 16);// tensor_dim0[31:16], tensor_dim1=1
        g1[3] = (uint32_t)(VOCAB & 0xFFFF) << 16;    // tile_dim0 = 32128  @ bits 127:112
        g1[4] = 0u;                                  // tile_dim1/tile_dim2 unused (1D tile)
        g1[5] = (uint32_t)VOCAB;                     // tensor_dim0_stride[31:0]
        g1[6] = 0u;                                  // stride high / dim1 stride
        g1[7] = 0u;

        asm volatile("tensor_store_from_lds %0, %1" :: "s"(g0), "s"(g1) : "memory");
        __builtin_amdgcn_s_wait_tensorcnt(0);        // keep LDS alive until DMA done
    }
}

extern "C" void kernel_launch(void* const* d_in, const int* in_sizes, int n_in,
                              void* d_out, int out_size, void* d_ws, size_t ws_size,
                              hipStream_t stream) {
    (void)in_sizes; (void)n_in; (void)d_ws; (void)ws_size; (void)out_size;
    const int* x  = (const int*)d_in[0];
    float* out    = (float*)d_out;
    row_hist_tdm_kernel<<<dim3(BATCH), dim3(TPB), 0, stream>>>(x, out);
}